// SpikingNeuralNet_90898687853116
// MI455X (gfx1250) — compile-verified
//
#include <hip/hip_runtime.h>
#include <cstdint>
#include <cstddef>

// ---------------------------------------------------------------------------
// Event-driven spiking net scan for MI455X (gfx1250, wave32).
// One workgroup (32 waves) owns the whole sequential T-loop; the CDNA5 async
// global->LDS pipe (ASYNCcnt) triple-buffers the per-step input row.
// ---------------------------------------------------------------------------

#define GLOBAL_AS __attribute__((address_space(1)))
#define LDS_AS    __attribute__((address_space(3)))

typedef float v4f __attribute__((ext_vector_type(4)));
typedef int   v4i __attribute__((vector_size(16)));   // matches builtin param type

static constexpr int   kT       = 2048;
static constexpr int   kN       = 4096;
static constexpr int   kThreads = 1024;          // 32 waves on one WGP
static constexpr int   kNPT     = kN / kThreads; // 4 contiguous neurons/thread
static constexpr float kVReset  = 1.0f;
static constexpr float kAlpha   = 0.01f;
static constexpr float kEps     = 1e-12f;
static constexpr int   kInf     = 0x7fffffff;

// --- CDNA5 async global->LDS copy (16B per lane) ---------------------------
__device__ __forceinline__ void async_copy_b128(const float* gsrc, float* ldst) {
#if __has_builtin(__builtin_amdgcn_global_load_async_to_lds_b128)
  __builtin_amdgcn_global_load_async_to_lds_b128(
      (GLOBAL_AS v4i*)const_cast<float*>(gsrc),
      (LDS_AS v4i*)ldst, 0, 0);
#else
  unsigned lds_off = (unsigned)(size_t)(LDS_AS void*)ldst;
  asm volatile("global_load_async_to_lds_b128 %0, %1, off"
               :: "v"(lds_off), "v"((unsigned long long)(size_t)gsrc)
               : "memory");
#endif
}

__device__ __forceinline__ void wait_async_le1() {
#if __has_builtin(__builtin_amdgcn_s_wait_asynccnt)
  __builtin_amdgcn_s_wait_asynccnt(1);
#else
  asm volatile("s_wait_asynccnt 1" ::: "memory");
#endif
}

__device__ __forceinline__ void wait_async_le0() {
#if __has_builtin(__builtin_amdgcn_s_wait_asynccnt)
  __builtin_amdgcn_s_wait_asynccnt(0);
#else
  asm volatile("s_wait_asynccnt 0" ::: "memory");
#endif
}

__global__ __launch_bounds__(kThreads, 1)
void snn_scan_kernel(const float* __restrict__ ts,        // [T]
                     const float* __restrict__ inp,       // [T,N]
                     const float* __restrict__ w,         // [N,N]
                     const float* __restrict__ v0,        // [N]
                     const float* __restrict__ i0,        // [N]
                     const float* __restrict__ mu,        // [2]
                     const float* __restrict__ s0u,       // [N]
                     const float* __restrict__ ru,        // [T,N]
                     const int*   __restrict__ max_spikes_p,
                     float* __restrict__ out)             // [T,N,3]
{
  __shared__ float lbuf[3][kN];   // triple-buffered input rows (48 KB of 320 KB)
  __shared__ int   fireSlot[3];   // rotating argmin slots (1 barrier/step)

  const int tid   = (int)threadIdx.x;
  const int nbase = tid * kNPT;

  const float dt    = ts[1] - ts[0];
  const float mu1   = mu[0];
  const float mu2   = mu[1];
  const float dtmu1 = dt * mu1;
  const float dtmu2 = dt * mu2;
  const int   max_spikes = max_spikes_p[0];

  // --- init state in registers --------------------------------------------
  float v[kNPT], cur[kNPT], s[kNPT];
  {
    v4f a = *(const v4f*)(v0  + nbase);
    v4f b = *(const v4f*)(i0  + nbase);
    v4f c = *(const v4f*)(s0u + nbase);
#pragma unroll
    for (int k = 0; k < kNPT; ++k) {
      v[k]   = a[k];
      cur[k] = b[k];
      s[k]   = __logf(c[k] + kEps) - kAlpha;
    }
  }
  if (tid < 3) fireSlot[tid] = kInf;

  // Prefetch rows 0 and 1; guarantee row 0 resident before the loop.
  async_copy_b128(inp + (size_t)0 * kN + nbase, &lbuf[0][nbase]);
  async_copy_b128(inp + (size_t)1 * kN + nbase, &lbuf[1][nbase]);
  wait_async_le1();
  __syncthreads();

  int nsp = 0;
  for (int t = 0; t < kT; ++t) {
    const int slot = t % 3;                        // lbuf & fireSlot index

    // Issue async prefetch of row t+2 (one full step of latency hiding).
    if (t + 2 < kT)
      async_copy_b128(inp + (size_t)(t + 2) * kN + nbase,
                      &lbuf[(t + 2) % 3][nbase]);
    // Reset the slot step t+1 will use (safe: last touched before barrier t-1).
    if (tid == 0) fireSlot[(t + 1) % 3] = kInf;

    // --- drift update (sigmoid uses OLD v, as in the reference) -----------
    v4f in4 = *(const v4f*)(&lbuf[slot][nbase]);
    float vn[kNPT], inx[kNPT], sn[kNPT];
    int cand = kInf;
#pragma unroll
    for (int k = 0; k < kNPT; ++k) {
      const float vo = v[k], io = cur[k];
      vn[k]  = fmaf(dtmu1, (io - vo) + in4[k], vo);     // v + dt*mu1*((i-v)+I)
      inx[k] = fmaf(-dtmu2, io, io);                    // i - dt*mu2*i
      const float sig = __builtin_amdgcn_rcpf(1.0f + __expf(-vo));
      sn[k]  = fmaf(dt, sig, s[k]);
      if (sn[k] >= 0.0f) cand = min(cand, nbase + k);   // lowest firing index
    }

    // --- wave32 min-reduction, then <=32 LDS atomics -----------------------
#pragma unroll
    for (int off = 16; off > 0; off >>= 1)
      cand = min(cand, __shfl_down(cand, off, 32));
    if ((tid & 31) == 0 && cand != kInf) atomicMin(&fireSlot[slot], cand);

    // Ensure row t+1 is resident (in-order async completion: leave only the
    // just-issued t+2 load outstanding).
    if (t + 2 < kT)      wait_async_le1();
    else if (t + 1 < kT) wait_async_le0();

    __syncthreads();

    const int  idx  = fireSlot[slot];
    const bool fire = (idx != kInf) & (nsp < max_spikes);   // wave-uniform

    if (fire) {
      const v4f wr = *(const v4f*)(w  + (size_t)idx * kN + nbase);
      const v4f r4 = *(const v4f*)(ru + (size_t)t   * kN + nbase);
#pragma unroll
      for (int k = 0; k < kNPT; ++k) {
        const bool ev = sn[k] >= 0.0f;
        v[k]   = ev ? (vn[k] - kVReset) : vn[k];
        cur[k] = ev ? inx[k] : (inx[k] + wr[k]);
        s[k]   = ev ? (__logf(r4[k] + kEps) - kAlpha) : sn[k];
      }
      ++nsp;
    } else {
#pragma unroll
      for (int k = 0; k < kNPT; ++k) { v[k] = vn[k]; cur[k] = inx[k]; s[k] = sn[k]; }
    }

    // --- write y_out[t] as 3 contiguous non-temporal b128 stores/lane ------
    float* ob = out + ((size_t)t * kN + nbase) * 3;   // 48B-aligned per lane
    v4f o0 = { v[0], cur[0], s[0], v[1] };
    v4f o1 = { cur[1], s[1], v[2], cur[2] };
    v4f o2 = { s[2], v[3], cur[3], s[3] };
    __builtin_nontemporal_store(o0, (v4f*)(ob + 0));
    __builtin_nontemporal_store(o1, (v4f*)(ob + 4));
    __builtin_nontemporal_store(o2, (v4f*)(ob + 8));
  }
}

extern "C" void kernel_launch(void* const* d_in, const int* in_sizes, int n_in,
                              void* d_out, int out_size, void* d_ws, size_t ws_size,
                              hipStream_t stream) {
  (void)in_sizes; (void)n_in; (void)out_size; (void)d_ws; (void)ws_size;
  const float* ts   = (const float*)d_in[0];
  const float* inp  = (const float*)d_in[1];
  const float* w    = (const float*)d_in[2];
  const float* v0   = (const float*)d_in[3];
  const float* i0   = (const float*)d_in[4];
  const float* mu   = (const float*)d_in[5];
  const float* s0u  = (const float*)d_in[6];
  const float* ru   = (const float*)d_in[7];
  const int*   msp  = (const int*)d_in[8];
  float* out = (float*)d_out;

  snn_scan_kernel<<<dim3(1), dim3(kThreads), 0, stream>>>(
      ts, inp, w, v0, i0, mu, s0u, ru, msp, out);
}